// AugmentedTransformer_9491877724454
// MI455X (gfx1250) — compile-verified
//
#include <hip/hip_runtime.h>
#include <hip/hip_bf16.h>
#include <math.h>

// ---- Problem constants (from reference setup_inputs) ----
#define B_   2
#define D_   32
#define N_   (B_ * D_)     // 64 batched matrices
#define C_   256
#define T_   64
#define TE_  1024
#define H_   8
#define CPH_ (C_ / H_)     // 32
#define GROUPS_ 32
#define CPG_ (C_ / GROUPS_) // 8 channels per group
#define EPS_ 1e-5f

typedef float v2f __attribute__((ext_vector_type(2)));
typedef float v8f __attribute__((ext_vector_type(8)));

// ---------------------------------------------------------------------------
// Core WMMA helper (compile-time strides so loads use immediate offsets):
// one 16x16 f32 D-tile = A(16xK) * B(Kx16) + acc via V_WMMA_F32_16X16X4_F32.
//  A fragment (ISA 16x4 f32 layout): lane L -> M = L%16, K = 2*(L/16)+{0,1}
//  B fragment (symmetric):           lane L -> N = L%16, K = 2*(L/16)+{0,1}
// ---------------------------------------------------------------------------
template <long A_M, long A_K, long B_K, long B_N, int KTOT>
__device__ __forceinline__ v8f wmma_tile_f32(const float* __restrict__ A,
                                             const float* __restrict__ Bm,
                                             v8f acc) {
  const int lane = threadIdx.x & 31;
  const int half = lane >> 4;       // 0: K pair {0,1}, 1: K pair {2,3}
  const int lr   = lane & 15;
  const float* ap = A  + (long)lr * A_M + (long)(2 * half) * A_K;
  const float* bp = Bm + (long)(2 * half) * B_K + (long)lr * B_N;
#pragma unroll 16
  for (int k = 0; k < KTOT; k += 4) {
    v2f a; a.x = ap[0]; a.y = ap[A_K];
    v2f b; b.x = bp[0]; b.y = bp[B_K];
    acc = __builtin_amdgcn_wmma_f32_16x16x4_f32(
        /*neg_a=*/false, a, /*neg_b=*/false, b,
        /*c_mod=*/(short)0, acc, /*reuse_a=*/false, /*reuse_b=*/false);
    ap += 4 * A_K;
    bp += 4 * B_K;
  }
  return acc;
}

// ---------------------------------------------------------------------------
// Kernel 1: GroupNorm.  One wave per (n, group); group = 8 chans x 64 t = 512.
// ---------------------------------------------------------------------------
__global__ void k_gn(const float* __restrict__ x,
                     const float* __restrict__ scale,
                     const float* __restrict__ bias,
                     float* __restrict__ h) {
  const int ng = blockIdx.x;          // 0 .. N_*GROUPS_-1
  const int n  = ng >> 5;
  const int g  = ng & 31;
  const long base = ((long)n * C_ + (long)g * CPG_) * T_;
  const float* xp = x + base;
  float s = 0.f, s2 = 0.f;
  for (int i = threadIdx.x; i < CPG_ * T_; i += 32) {
    float v = xp[i];
    s += v; s2 += v * v;
  }
#pragma unroll
  for (int o = 16; o > 0; o >>= 1) {
    s  += __shfl_xor(s,  o, 32);
    s2 += __shfl_xor(s2, o, 32);
  }
  const float mean = s * (1.f / (CPG_ * T_));
  const float var  = s2 * (1.f / (CPG_ * T_)) - mean * mean;
  const float inv  = rsqrtf(var + EPS_);
  float* hp = h + base;
  for (int i = threadIdx.x; i < CPG_ * T_; i += 32) {
    const int c = g * CPG_ + (i >> 6);   // i / T_
    hp[i] = (xp[i] - mean) * inv * scale[c] + bias[c];
  }
}

// ---------------------------------------------------------------------------
// Kernel 2: Generic batched GEMM (compile-time shape)
//   D[b] = A * B[b] (+bias[m]) (+add[b])
// A: MxK row-major (shared across batch). B: KxN row-major.
// One 16x16 tile per wave, 8 waves per block. gridDim.y = batch.
// ---------------------------------------------------------------------------
template <int M, int N, int K, bool HAS_BIAS, bool HAS_ADD>
__global__ void k_gemm(const float* __restrict__ A,
                       const float* __restrict__ Bm,
                       const float* __restrict__ bias,
                       const float* __restrict__ add,
                       float* __restrict__ Dm) {
  constexpr int NT_N = N / 16;
  const int batch = blockIdx.y;
  const int wave  = threadIdx.x >> 5;
  const int tile  = blockIdx.x * 8 + wave;
  const int tm = tile / NT_N;
  const int tn = tile % NT_N;
  const int lane = threadIdx.x & 31;
  const int half = lane >> 4;
  const int lr   = lane & 15;

  v8f acc = {};
  acc = wmma_tile_f32<(long)K, 1L, (long)N, 1L, K>(
      A + (long)tm * 16 * K,
      Bm + (long)batch * K * N + tn * 16, acc);

  const long dbase = (long)batch * M * N;
#pragma unroll
  for (int j = 0; j < 8; ++j) {
    const int m = tm * 16 + j + 8 * half;
    const int n = tn * 16 + lr;
    float v = acc[j];
    if constexpr (HAS_BIAS) v += bias[m];
    if constexpr (HAS_ADD)  v += add[dbase + (long)m * N + n];
    Dm[dbase + (long)m * N + n] = v;
  }
}

// ---------------------------------------------------------------------------
// Kernel 3: QK^T scores per (n, head).  S[t,s] = (1/sqrt(cph)) sum_ch q*k.
// A[M=t,K=ch] = q[ch,t]  (A_M = 1, A_K = H_*T_); B[K=ch,N=s] similarly.
// One block (8 waves) per (n,h); 16 tiles -> 2 tiles per wave.
// ---------------------------------------------------------------------------
__global__ void k_qk(const float* __restrict__ qkv,
                     float* __restrict__ scores) {
  const int nh = blockIdx.x;           // n*H_ + h
  const int n  = nh >> 3;
  const int h  = nh & 7;
  const int wave = threadIdx.x >> 5;
  const int lane = threadIdx.x & 31;
  const int half = lane >> 4;
  const int lr   = lane & 15;

  const float* qb = qkv + ((long)n * 3 * C_ + h) * T_;          // o = ch*H + h
  const float* kb = qkv + ((long)n * 3 * C_ + C_ + h) * T_;     // o = C + ch*H + h
  const float sc2 = 0.17677669529663687f;                       // 1/sqrt(32)

#pragma unroll
  for (int r = 0; r < 2; ++r) {
    const int tile = wave * 2 + r;
    const int tt = tile >> 2;      // t tile (0..3)
    const int ts = tile & 3;       // s tile (0..3)
    v8f acc = {};
    acc = wmma_tile_f32<1L, (long)(H_ * T_), (long)(H_ * T_), 1L, CPH_>(
        qb + tt * 16, kb + ts * 16, acc);
    const long obase = ((long)nh * T_ + tt * 16) * T_ + ts * 16;
#pragma unroll
    for (int j = 0; j < 8; ++j) {
      const int m = j + 8 * half;
      scores[obase + (long)m * T_ + lr] = acc[j] * sc2;
    }
  }
}

// ---------------------------------------------------------------------------
// Kernel 4: rel-pos embedding + ReLU staged in LDS, then wa = w_aug3*relu(emb).
// One block per (b, t, s-tile of 16).  The B strip (256 K-values x 16 cols)
// is staged TRANSPOSED in LDS (Bs[col][K], row stride 260 => banks shift by 4)
// so each lane's K-pair is one ds_load_b64, conflict-free across the wave.
// Each of 8 waves computes two 16x16 output tiles (full 256 output channels).
// ---------------------------------------------------------------------------
#define BS_STRIDE 260   // 256 + 4 pad: stride % 64 == 4 -> 64 distinct banks/b64
__global__ void k_wa(const int* __restrict__ frame_indices,
                     const float* __restrict__ w_aug1,
                     const float* __restrict__ b_aug1,
                     const float* __restrict__ tembproj,
                     const float* __restrict__ w_aug3,
                     const float* __restrict__ b_aug3,
                     float* __restrict__ wa) {
  const int s0 = blockIdx.x * 16;
  const int t  = blockIdx.y;
  const int b  = blockIdx.z;
  const int tid = threadIdx.x;

  __shared__ float r3[3][16];
  __shared__ float Bs[16][BS_STRIDE];   // relu(emb)^T strip, ~16.6 KB

  if (tid < 16) {
    const int s = s0 + tid;
    const int rel = frame_indices[b * T_ + s] - frame_indices[b * T_ + t];
    const float rp = rel > 0 ? (float)rel : 0.f;
    const float rn = rel < 0 ? (float)(-rel) : 0.f;
    r3[0][tid] = log1pf(rp);
    r3[1][tid] = log1pf(rn);
    r3[2][tid] = (rel == 0) ? 0.6931471805599453f : 0.f;   // log1p(1)
  }
  __syncthreads();

  {
    const int i = tid;                      // channel (K index) 0..255
    const float w0 = w_aug1[i * 3 + 0];
    const float w1 = w_aug1[i * 3 + 1];
    const float w2 = w_aug1[i * 3 + 2];
    const float base = b_aug1[i] + tembproj[((long)b * C_ + i) * T_ + t];
#pragma unroll
    for (int j = 0; j < 16; ++j) {
      float e = fmaf(w0, r3[0][j], fmaf(w1, r3[1][j], fmaf(w2, r3[2][j], base)));
      Bs[j][i] = fmaxf(e, 0.f);
    }
  }
  __syncthreads();

  const int wave = tid >> 5;
  const int lane = tid & 31;
  const int half = lane >> 4;
  const int lr   = lane & 15;
  const int kb   = 2 * half;
  const float* brow = &Bs[lr][kb];          // lane's column, K-contiguous

#pragma unroll
  for (int r = 0; r < 2; ++r) {
    const int tm = wave * 2 + r;            // output-channel tile (0..15)
    v8f acc = {};
    const float* ap = w_aug3 + (long)(tm * 16 + lr) * C_ + kb;
#pragma unroll 16
    for (int k = 0; k < C_; k += 4) {
      v2f a; a.x = ap[k];     a.y = ap[k + 1];       // contiguous -> b64
      v2f bb; bb.x = brow[k]; bb.y = brow[k + 1];    // contiguous -> ds b64
      acc = __builtin_amdgcn_wmma_f32_16x16x4_f32(
          false, a, false, bb, (short)0, acc, false, false);
    }
#pragma unroll
    for (int j = 0; j < 8; ++j) {
      const int o = tm * 16 + j + 8 * half;
      wa[(((long)b * C_ + o) * T_ + t) * T_ + s0 + lr] = acc[j] + b_aug3[o];
    }
  }
}

// ---------------------------------------------------------------------------
// Kernel 5: per-(n,c) attention: online softmax over s + weighted sum with v.
// logits[t,s] = scores[n, c%H, t, s] + wa[b, c, t, s];  hv[t] = softmax . v
// One block per (n,c); 64 threads, one output row t per thread.
// ---------------------------------------------------------------------------
__global__ void k_attn(const float* __restrict__ scores,
                       const float* __restrict__ wa,
                       const float* __restrict__ qkv,
                       const float* __restrict__ attn_mask,
                       float* __restrict__ hv) {
  const int bc = blockIdx.x;          // n*C_ + c
  const int n  = bc >> 8;
  const int c  = bc & (C_ - 1);
  const int b  = n >> 5;              // n / D_
  const int h  = c & (H_ - 1);
  const int t  = threadIdx.x;

  const float* srow = scores + (((long)(n * H_ + h)) * T_ + t) * T_;
  const float* wrow = wa + (((long)b * C_ + c) * T_ + t) * T_;
  const float* vrow = qkv + ((long)n * 3 * C_ + 2 * C_ + c) * T_;  // v chan c
  const float* mrow = attn_mask + b * T_;

  float m = -INFINITY, l = 0.f, acc = 0.f;
  for (int s = 0; s < T_; ++s) {
    if (mrow[s] > 0.f) {
      const float xv = srow[s] + wrow[s];
      const float nm = fmaxf(m, xv);
      const float rs = __expf(m - nm);
      const float p  = __expf(xv - nm);
      l   = l * rs + p;
      acc = acc * rs + p * vrow[s];
      m = nm;
    }
  }
  hv[((long)n * C_ + c) * T_ + t] = acc / l;
}

// ---------------------------------------------------------------------------
// Launch
// ---------------------------------------------------------------------------
extern "C" void kernel_launch(void* const* d_in, const int* in_sizes, int n_in,
                              void* d_out, int out_size, void* d_ws, size_t ws_size,
                              hipStream_t stream) {
  const float* x       = (const float*)d_in[0];
  const float* temb    = (const float*)d_in[1];
  const int*   fidx    = (const int*)  d_in[2];
  const float* amask   = (const float*)d_in[3];
  const float* nscale  = (const float*)d_in[4];
  const float* nbias   = (const float*)d_in[5];
  const float* w_qkv   = (const float*)d_in[6];
  const float* b_qkv   = (const float*)d_in[7];
  const float* w_aug1  = (const float*)d_in[8];
  const float* b_aug1  = (const float*)d_in[9];
  const float* w_aug2  = (const float*)d_in[10];
  const float* b_aug2  = (const float*)d_in[11];
  const float* w_aug3  = (const float*)d_in[12];
  const float* b_aug3  = (const float*)d_in[13];
  const float* w_proj  = (const float*)d_in[14];
  const float* b_proj  = (const float*)d_in[15];
  float* out = (float*)d_out;

  // Workspace layout (floats). hv reuses h (h dead after qkv GEMM).
  float* ws = (float*)d_ws;
  float* h        = ws;                       // N*C*T      = 1,048,576
  float* hv       = ws;                       // reuse of h
  float* qkv      = ws + 1048576;             // N*3C*T     = 3,145,728
  float* tembproj = ws + 4194304;             // B*C*T      =    32,768
  float* wa       = ws + 4227072;             // B*C*T*T    = 2,097,152
  float* scores   = ws + 6324224;             // N*H*T*T    = 2,097,152
  // total: 8,421,376 floats = ~33.7 MB

  // 1) GroupNorm
  k_gn<<<dim3(N_ * GROUPS_), dim3(32), 0, stream>>>(x, nscale, nbias, h);

  // 2) qkv = w_qkv (768x256) @ h_n (256x64) + b_qkv      [per n]
  //    tiles = 48*4 = 192 -> 24 blocks of 8 waves
  k_gemm<3 * C_, T_, C_, true, false><<<dim3(24, N_), dim3(256), 0, stream>>>(
      w_qkv, h, b_qkv, nullptr, qkv);

  // 3) tembproj = w_aug2 (256x1024) @ temb_b (1024x64) + b_aug2   [per b]
  //    tiles = 16*4 = 64 -> 8 blocks of 8 waves
  k_gemm<C_, T_, TE_, true, false><<<dim3(8, B_), dim3(256), 0, stream>>>(
      w_aug2, temb, b_aug2, nullptr, tembproj);

  // 4) per-head scores
  k_qk<<<dim3(N_ * H_), dim3(256), 0, stream>>>(qkv, scores);

  // 5) wa = w_aug3 @ relu(emb)   [per (b, t, s-tile)]
  k_wa<<<dim3(T_ / 16, T_, B_), dim3(256), 0, stream>>>(
      fidx, w_aug1, b_aug1, tembproj, w_aug3, b_aug3, wa);

  // 6) attention (online softmax) -> hv
  k_attn<<<dim3(N_ * C_), dim3(T_), 0, stream>>>(scores, wa, qkv, amask, hv);

  // 7) out = x + w_proj @ hv + b_proj   [per n]
  k_gemm<C_, T_, C_, true, true><<<dim3(8, N_), dim3(256), 0, stream>>>(
      w_proj, hv, b_proj, x, out);
}